// NodeLevelAttentionLayer_65910568124771
// MI455X (gfx1250) — compile-verified
//
#include <hip/hip_runtime.h>
#include <hip/hip_bf16.h>

typedef __attribute__((ext_vector_type(2))) float v2f;
typedef __attribute__((ext_vector_type(8))) float v8f;

#define LRELU_SLOPE 0.01f

__device__ __forceinline__ float lrelu_f(float x) {
    return x >= 0.0f ? x : LRELU_SLOPE * x;
}

// ---------------------------------------------------------------------------
// Generic fp32 WMMA GEMM: C[16x16 tile] = A(MxK) * B(KxN), one wave per tile.
// Grid: (M/16, N/16), block: 32 threads (one wave32).
// A operand layout (V_WMMA_F32_16X16X4_F32, 32-bit A 16x4):
//   lane&15 = M row; (lane>>4)*2 = K base; VGPR0/1 = K, K+1
// B operand (4x16): VGPR0 = row (k + (lane>>4)*2), VGPR1 = next row, col=lane&15
// C/D (16x16): VGPR r -> row r + (lane>>4)*8, col = lane&15
// ---------------------------------------------------------------------------
__global__ void wmma_gemm_f32(const float* __restrict__ A,
                              const float* __restrict__ B,
                              float* __restrict__ C,
                              int K, int lda, int ldb, int ldc,
                              int apply_lrelu) {
    const int lane = threadIdx.x;
    const int i0 = blockIdx.x * 16;
    const int j0 = blockIdx.y * 16;
    const int m  = lane & 15;
    const int kh = (lane >> 4) * 2;

    const float* ap = A + (size_t)(i0 + m) * lda + kh;
    const float* bp = B + (size_t)kh * ldb + j0 + m;

    v8f c = {};
    for (int k = 0; k < K; k += 4) {
        v2f a, b;
        a.x = ap[0];
        a.y = ap[1];
        b.x = bp[0];
        b.y = bp[ldb];
        c = __builtin_amdgcn_wmma_f32_16x16x4_f32(false, a, false, b,
                                                  (short)0, c, false, false);
        ap += 4;
        bp += (size_t)4 * ldb;
    }
    const int rbase = (lane >> 4) * 8;
    for (int r = 0; r < 8; ++r) {
        float v = c[r];
        if (apply_lrelu) v = lrelu_f(v);
        C[(size_t)(i0 + rbase + r) * ldc + j0 + m] = v;
    }
}

// ---------------------------------------------------------------------------
// Row projections: aself = h@pvn[:128], aneigh = h@pvn[128:], bself = h@pve[:128]
// ---------------------------------------------------------------------------
__global__ void proj_node(const float* __restrict__ h,
                          const float* __restrict__ pvn,
                          const float* __restrict__ pve,
                          float* __restrict__ aself,
                          float* __restrict__ aneigh,
                          float* __restrict__ bself) {
    int i = blockIdx.x * blockDim.x + threadIdx.x;
    float s0 = 0.f, s1 = 0.f, s2 = 0.f;
    const float* row = h + (size_t)i * 128;
    for (int k = 0; k < 128; ++k) {
        float hv = row[k];
        s0 += hv * pvn[k];
        s1 += hv * pvn[128 + k];
        s2 += hv * pve[k];
    }
    aself[i]  = s0;
    aneigh[i] = s1;
    bself[i]  = s2;
}

__global__ void proj_edge(const float* __restrict__ he,
                          const float* __restrict__ pve,
                          float* __restrict__ eproj) {
    int j = blockIdx.x * blockDim.x + threadIdx.x;
    float s = 0.f;
    const float* row = he + (size_t)j * 64;
    for (int k = 0; k < 64; ++k) s += row[k] * pve[128 + k];
    eproj[j] = s;
}

// ---------------------------------------------------------------------------
// Masked row statistics over mask^T:  for output row i,
//   rowmax[i] = max_{j: mask[j][i]>0} proj[j]   (raw, -1e30 if empty)
//   deg[i]    = #{j: mask[j][i]>0}
// Block: 256 threads = 64 i-columns x 4 j-strides; coalesced 256B mask reads.
// ---------------------------------------------------------------------------
__global__ void rowstat_kernel(const int* __restrict__ mask,
                               const float* __restrict__ proj,
                               int J, int ldm,
                               float* __restrict__ rowmax,
                               float* __restrict__ deg) {
    const int tx = threadIdx.x & 63;
    const int ty = threadIdx.x >> 6;       // 0..3
    const int i  = blockIdx.x * 64 + tx;

    const int* mp = mask + (size_t)ty * ldm + i;
    float mx = -1e30f;
    int cnt = 0;
    for (int j = ty; j < J; j += 4) {
        __builtin_prefetch(mp + (size_t)32 * ldm, 0, 1);
        if (*mp > 0) {
            mx = fmaxf(mx, proj[j]);
            ++cnt;
        }
        mp += (size_t)4 * ldm;
    }
    __shared__ float smx[4][64];
    __shared__ int   scn[4][64];
    smx[ty][tx] = mx;
    scn[ty][tx] = cnt;
    __syncthreads();
    if (ty == 0) {
        for (int q = 1; q < 4; ++q) {
            mx = fmaxf(mx, smx[q][tx]);
            cnt += scn[q][tx];
        }
        rowmax[i] = mx;
        deg[i]    = (float)cnt;
    }
}

// ---------------------------------------------------------------------------
// Fused masked-softmax aggregation (the big GEMM):
//   out[i,:] = ( sum_j w_ij * feat[j,:] ) / ( sum_j w_ij * deg_i )
//   w_ij = mask[j][i]>0 ? exp( lrelu(aself_i + proj_j) - smax_i ) : 0,
//   smax_i = lrelu(aself_i + rowmax_i)   (exact masked row max, monotonicity)
//
// One block (256 threads = 8 waves) per 16-row i-block. The K dimension is
// split across the 8 waves (each wave owns k = 4*wave + 32*n); every wave
// generates its 16x4 attention-weight A tile in registers and feeds
// V_WMMA_F32_16X16X4_F32 against feature-row B tiles. Per-wave fp32
// accumulators are then reduced through LDS (8 x 16 x F slab) and scaled.
// ---------------------------------------------------------------------------
template <int FT>
__global__ __launch_bounds__(256)
void agg_kernel(const int* __restrict__ mask,
                const float* __restrict__ aself,
                const float* __restrict__ proj,
                const float* __restrict__ rowmax,
                const float* __restrict__ deg,
                const float* __restrict__ feat,
                float* __restrict__ outp,
                int J, int ldm) {
    constexpr int F  = FT * 16;
    constexpr int NW = 8;                      // waves per block
    const int tid  = threadIdx.x;
    const int wave = tid >> 5;
    const int lane = tid & 31;
    const int i0 = blockIdx.x * 16;
    const int m  = lane & 15;
    const int kh = (lane >> 4) * 2;

    const float as_i = aself[i0 + m];
    const float smax = lrelu_f(as_i + rowmax[i0 + m]);

    v8f c[FT];
    for (int t = 0; t < FT; ++t) c[t] = (v8f){};
    float wsum = 0.0f;

    // per-wave strided K walk: j0 = 4*wave + kh + 32*n
    const int kstart = wave * 4;
    const int*   mp = mask + (size_t)(kstart + kh) * ldm + i0 + m;
    const float* pp = proj + kstart + kh;
    const float* fp = feat + (size_t)(kstart + kh) * F + m;

    for (int k = kstart; k < J; k += 32) {
        __builtin_prefetch(mp + (size_t)64 * ldm, 0, 1);
        const int a0 = mp[0];
        const int a1 = mp[ldm];
        const float p0 = pp[0];
        const float p1 = pp[1];
        float w0 = (a0 > 0) ? expf(lrelu_f(as_i + p0) - smax) : 0.0f;
        float w1 = (a1 > 0) ? expf(lrelu_f(as_i + p1) - smax) : 0.0f;
        wsum += w0 + w1;
        v2f a;
        a.x = w0;
        a.y = w1;
        for (int t = 0; t < FT; ++t) {
            v2f b;
            b.x = fp[t * 16];
            b.y = fp[F + t * 16];
            c[t] = __builtin_amdgcn_wmma_f32_16x16x4_f32(false, a, false, b,
                                                         (short)0, c[t],
                                                         false, false);
        }
        mp += (size_t)32 * ldm;
        pp += 32;
        fp += (size_t)32 * F;
    }

    // ---- cross-wave reduction through LDS ----
    __shared__ float sacc[NW * 16 * F];        // per-wave partial C tiles
    __shared__ float swsum[NW][16];            // per-wave partial w row sums

    float wtot = wsum + __shfl_xor(wsum, 16, 32);   // fold lane halves
    if (lane < 16) swsum[wave][lane] = wtot;

    const int rbase = (lane >> 4) * 8;
    for (int r = 0; r < 8; ++r) {
        const int row = rbase + r;
        for (int t = 0; t < FT; ++t) {
            sacc[((size_t)wave * 16 + row) * F + t * 16 + m] = c[t][r];
        }
    }
    __syncthreads();

    for (int idx = tid; idx < 16 * F; idx += 256) {
        const int row = idx / F;
        const int col = idx - row * F;
        float acc = 0.0f, wt = 0.0f;
        for (int w = 0; w < NW; ++w) {
            acc += sacc[((size_t)w * 16 + row) * F + col];
            wt  += swsum[w][row];
        }
        const int gi = i0 + row;
        outp[(size_t)gi * F + col] = acc / (wt * deg[gi]);
    }
}

// ---------------------------------------------------------------------------
// Launch
// ---------------------------------------------------------------------------
extern "C" void kernel_launch(void* const* d_in, const int* in_sizes, int n_in,
                              void* d_out, int out_size, void* d_ws, size_t ws_size,
                              hipStream_t stream) {
    constexpr int N = 4096, E = 16384;
    constexpr int Fn = 128, Fe = 64, Dn = 128, De = 64;

    const float* Xn  = (const float*)d_in[0];   // N x Fn
    const float* Xe  = (const float*)d_in[1];   // E x Fe
    const int*   Adj = (const int*)d_in[2];     // N x N
    const int*   EAdj= (const int*)d_in[3];     // E x N
    const float* Wn  = (const float*)d_in[4];   // Fn x Dn
    const float* We  = (const float*)d_in[5];   // Fe x De
    const float* pvn = (const float*)d_in[6];   // 2*Dn
    const float* pve = (const float*)d_in[7];   // Dn + De

    float* ws = (float*)d_ws;
    float* h       = ws;                 ws += (size_t)N * Dn;   // 4096x128
    float* he      = ws;                 ws += (size_t)E * De;   // 16384x64
    float* aself   = ws;                 ws += N;
    float* aneigh  = ws;                 ws += N;
    float* bself   = ws;                 ws += N;
    float* eproj   = ws;                 ws += E;
    float* mN      = ws;                 ws += N;
    float* degN    = ws;                 ws += N;
    float* mE      = ws;                 ws += N;
    float* degE    = ws;                 ws += N;
    float* meansN  = ws;                 ws += (size_t)N * Fn;   // 4096x128
    float* meansE  = ws;                 ws += (size_t)N * Fe;   // 4096x64

    float* out = (float*)d_out;          // N x (Dn + De) = 4096 x 192

    // 1) h = Xn @ Wn ; he = Xe @ We
    wmma_gemm_f32<<<dim3(N / 16, Dn / 16), 32, 0, stream>>>(Xn, Wn, h,
                                                            Fn, Fn, Dn, Dn, 0);
    wmma_gemm_f32<<<dim3(E / 16, De / 16), 32, 0, stream>>>(Xe, We, he,
                                                            Fe, Fe, De, De, 0);

    // 2) projections
    proj_node<<<N / 256, 256, 0, stream>>>(h, pvn, pve, aself, aneigh, bself);
    proj_edge<<<E / 256, 256, 0, stream>>>(he, pve, eproj);

    // 3) masked row max + degree over mask^T
    rowstat_kernel<<<N / 64, 256, 0, stream>>>(Adj,  aneigh, N, N, mN, degN);
    rowstat_kernel<<<N / 64, 256, 0, stream>>>(EAdj, eproj,  E, N, mE, degE);

    // 4) fused softmax-weighted aggregation (big WMMA GEMMs, 8 waves/block)
    agg_kernel<8><<<N / 16, 256, 0, stream>>>(Adj,  aself, aneigh, mN, degN,
                                              Xn, meansN, N, N);
    agg_kernel<4><<<N / 16, 256, 0, stream>>>(EAdj, bself, eproj,  mE, degE,
                                              Xe, meansE, E, N);

    // 5) output projections with leaky-relu, written into concatenated out
    wmma_gemm_f32<<<dim3(N / 16, Dn / 16), 32, 0, stream>>>(meansN, Wn, out,
                                                            Fn, Fn, Dn,
                                                            Dn + De, 1);
    wmma_gemm_f32<<<dim3(N / 16, De / 16), 32, 0, stream>>>(meansE, We,
                                                            out + Dn,
                                                            Fe, Fe, De,
                                                            Dn + De, 1);
}